// BilateralFilter_19585050869993
// MI455X (gfx1250) — compile-verified
//
#include <hip/hip_runtime.h>
#include <hip/hip_bf16.h>

// Bilateral filter, 1x3x128x128 fp32, k=35, sigma=5.6, reflect padding.
// Normalizations cancel: out_c = sum(w * p_c) / sum(w), w = exp(-dd^2/2s^2)*g(di)*g(dj)
// dd = S/3 with S = |dr|+|dg|+|db|  =>  range weight = exp2(S^2 * C2),
// C2 = -log2(e) / (9 * 2 * sigma^2)  (mean-fold + base-2 fold, native v_exp_f32)

#define KSIZE   35
#define PAD     17
#define TILE    4
#define HALO    (TILE + KSIZE - 1)      // 38
#define IMG     128
#define SIGMA_F 5.6f
#define LOG2E   1.44269504088896340736f
#define INV2S2  (1.0f / (2.0f * SIGMA_F * SIGMA_F))
#define C2      (-LOG2E / (9.0f * 2.0f * SIGMA_F * SIGMA_F))

#if __has_builtin(__builtin_amdgcn_exp2f)
#define EXP2F(v) __builtin_amdgcn_exp2f(v)   // direct v_exp_f32
#else
#define EXP2F(v) exp2f(v)
#endif

__global__ __launch_bounds__(256) void bilateral35_kernel(const float* __restrict__ x,
                                                          float* __restrict__ out) {
    __shared__ float tile[3][HALO][HALO];   // 17328 B
    __shared__ float gtab[KSIZE];           // 1-D spatial gaussian

    const int tid = threadIdx.x;
    const int bx  = blockIdx.x;             // tile col
    const int by  = blockIdx.y;             // tile row
    const int ti0 = by * TILE - PAD;
    const int tj0 = bx * TILE - PAD;

    // 1-D spatial gaussian table (separable)
    if (tid < KSIZE) {
        float d = (float)(tid - PAD);
        gtab[tid] = EXP2F(-d * d * (INV2S2 * LOG2E));
    }

    // ---- stage halo tile (reflect padding) into LDS via gfx1250 async DMA ----
    const int NELEM = 3 * HALO * HALO;      // 4332
#if defined(__gfx1250__) && __has_builtin(__builtin_amdgcn_global_load_async_to_lds_b32) && \
    __has_builtin(__builtin_amdgcn_s_wait_asynccnt)
    {
        typedef __attribute__((address_space(1))) int g_i32;   // global int
        typedef __attribute__((address_space(3))) int l_i32;   // LDS int
        l_i32* lbase = (l_i32*)&tile[0][0][0];
        for (int idx = tid; idx < NELEM; idx += 256) {
            int c   = idx / (HALO * HALO);
            int rem = idx - c * (HALO * HALO);
            int r   = rem / HALO;
            int cc  = rem - r * HALO;
            int gi = ti0 + r;  gi = gi < 0 ? -gi : (gi > IMG - 1 ? 2 * IMG - 2 - gi : gi);
            int gj = tj0 + cc; gj = gj < 0 ? -gj : (gj > IMG - 1 ? 2 * IMG - 2 - gj : gj);
            g_i32* gp = (g_i32*)&x[(c * IMG + gi) * IMG + gj];
            __builtin_amdgcn_global_load_async_to_lds_b32(gp, lbase + idx, 0, 0);
        }
        __builtin_amdgcn_s_wait_asynccnt(0);
    }
#else
    {
        float* lbase = &tile[0][0][0];
        for (int idx = tid; idx < NELEM; idx += 256) {
            int c   = idx / (HALO * HALO);
            int rem = idx - c * (HALO * HALO);
            int r   = rem / HALO;
            int cc  = rem - r * HALO;
            int gi = ti0 + r;  gi = gi < 0 ? -gi : (gi > IMG - 1 ? 2 * IMG - 2 - gi : gi);
            int gj = tj0 + cc; gj = gj < 0 ? -gj : (gj > IMG - 1 ? 2 * IMG - 2 - gj : gj);
            lbase[idx] = x[(c * IMG + gi) * IMG + gj];
        }
    }
#endif
    __syncthreads();

    // ---- each pixel handled by 16 contiguous lanes (same wave half) ----
    const int pixel = tid >> 4;             // 0..15 -> 4x4 tile position
    const int s     = tid & 15;             // window-row slice 0..15
    const int pi    = pixel >> 2;
    const int pj    = pixel & 3;

    const float r0 = tile[0][PAD + pi][PAD + pj];
    const float g0 = tile[1][PAD + pi][PAD + pj];
    const float b0 = tile[2][PAD + pi][PAD + pj];

    float ws = 0.0f, ar = 0.0f, ag = 0.0f, ab = 0.0f;

    // rows s, s+16, s+32 (35 = 2*16 + 3) -> all rows covered exactly once
    for (int di = s; di < KSIZE; di += 16) {
        const float grow = gtab[di];
        const float* rowR = &tile[0][pi + di][pj];
        const float* rowG = &tile[1][pi + di][pj];
        const float* rowB = &tile[2][pi + di][pj];
        float pws = 0.0f, par = 0.0f, pag = 0.0f, pab = 0.0f;
        #pragma unroll 7
        for (int dj = 0; dj < KSIZE; ++dj) {
            float r = rowR[dj], g = rowG[dj], b = rowB[dj];
            // |.| folds into v_add_f32 source modifiers
            float S = fabsf(r - r0) + fabsf(g - g0) + fabsf(b - b0);
            float w = EXP2F(S * S * C2) * gtab[dj];     // native v_exp_f32
            pws += w;
            par = fmaf(w, r, par);
            pag = fmaf(w, g, pag);
            pab = fmaf(w, b, pab);
        }
        // fold row gaussian once per row instead of once per tap
        ws = fmaf(pws, grow, ws);
        ar = fmaf(par, grow, ar);
        ag = fmaf(pag, grow, ag);
        ab = fmaf(pab, grow, ab);
    }

    // wave32 butterfly reduction over the 16 slice lanes (masks < 16 stay in-group)
    #pragma unroll
    for (int m = 8; m >= 1; m >>= 1) {
        ws += __shfl_xor(ws, m, 32);
        ar += __shfl_xor(ar, m, 32);
        ag += __shfl_xor(ag, m, 32);
        ab += __shfl_xor(ab, m, 32);
    }

    if (s == 0) {
        const int gi = by * TILE + pi;
        const int gj = bx * TILE + pj;
        const float inv = 1.0f / ws;
        out[(0 * IMG + gi) * IMG + gj] = ar * inv;
        out[(1 * IMG + gi) * IMG + gj] = ag * inv;
        out[(2 * IMG + gi) * IMG + gj] = ab * inv;
    }
}

extern "C" void kernel_launch(void* const* d_in, const int* in_sizes, int n_in,
                              void* d_out, int out_size, void* d_ws, size_t ws_size,
                              hipStream_t stream) {
    const float* x = (const float*)d_in[0];
    float* out     = (float*)d_out;
    dim3 grid(IMG / TILE, IMG / TILE);      // 32 x 32 = 1024 workgroups
    bilateral35_kernel<<<grid, 256, 0, stream>>>(x, out);
}